// TorchInnerProductCost_9577777070666
// MI455X (gfx1250) — compile-verified
//
#include <hip/hip_runtime.h>

typedef __attribute__((ext_vector_type(2))) float v2f;
typedef __attribute__((ext_vector_type(8))) float v8f;

namespace {
constexpr int kC = 128;
constexpr int kH = 160;
constexpr int kW = 320;
constexpr int kD = 48;
constexpr unsigned kHW = (unsigned)kH * (unsigned)kW;   // 51200 elements
}

// Banded-GEMM cost volume via V_WMMA_F32_16X16X4_F32.
// One wave per 16-wide w-tile; 4 w'-tiles (diagonal band d in [0,48)).
__global__ __launch_bounds__(128) void corr_volume_wmma(
    const float* __restrict__ left,
    const float* __restrict__ right,
    float* __restrict__ out)
{
  const int lane = threadIdx.x & 31;
  const int wave = threadIdx.x >> 5;
  const int i    = (int)blockIdx.x * 4 + wave;   // w-tile index 0..19
  const int h    = (int)blockIdx.y;
  const int n    = (int)blockIdx.z;

  const int laneN = lane & 15;   // M (for A) / N (for B) within the tile
  const int laneG = lane >> 4;   // 0: K=0,1   1: K=2,3

  const unsigned slice = (unsigned)n * kC * kHW + (unsigned)h * kW;

  // A fragment per K-step: a.x = left[c0+2g][16i+laneN], a.y = next channel
  unsigned offA = slice + (unsigned)(laneG * 2) * kHW + (unsigned)(16 * i + laneN);

  unsigned offB[4];
  bool valid[4];
#pragma unroll
  for (int jj = 0; jj < 4; ++jj) {
    const int j  = i - 3 + jj;          // w'-tile index (may be < 0 at left edge)
    valid[jj]    = (j >= 0);            // per-tile uniform: tiles are 16-aligned
    const int jc = (j < 0) ? 0 : j;     // clamp: harmless loads, result discarded
    offB[jj] = slice + (unsigned)(laneG * 2) * kHW + (unsigned)(16 * jc + laneN);
  }

  v8f acc[4] = {};

  // K loop over channels, 4 at a time (32 iterations, 4 WMMAs each).
#pragma unroll 2
  for (int kk = 0; kk < kC / 4; ++kk) {
    v2f a;
    a.x = left[offA];
    a.y = left[offA + kHW];
    offA += 4u * kHW;
#pragma unroll
    for (int jj = 0; jj < 4; ++jj) {
      v2f b;
      b.x = right[offB[jj]];
      b.y = right[offB[jj] + kHW];
      offB[jj] += 4u * kHW;
      // D = A(16x4) * B(4x16) + C ; fp32 throughout (matches reference precision)
      acc[jj] = __builtin_amdgcn_wmma_f32_16x16x4_f32(
          /*neg_a=*/false, a, /*neg_b=*/false, b,
          /*c_mod=*/(short)0, acc[jj],
          /*reuse_a=*/false, /*reuse_b=*/false);
    }
  }

  // Scatter D tiles into vol[n][d][h][w]; d = 16*(i-j) + M - N.
  // Union over jj covers each d in [0,48) exactly once; invalid (edge) tiles
  // contribute the required zero padding.
  const unsigned outSlab = (unsigned)n * kD * kHW + (unsigned)h * kW;
#pragma unroll
  for (int jj = 0; jj < 4; ++jj) {
    const int dBase = 16 * (3 - jj);
#pragma unroll
    for (int r = 0; r < 8; ++r) {
      const int M = r + 8 * laneG;
      const int d = dBase + M - laneN;
      if ((unsigned)d < (unsigned)kD) {
        const unsigned idx = outSlab + (unsigned)d * kHW + (unsigned)(16 * i + M);
        const float v = valid[jj] ? acc[jj][r] : 0.0f;
        __builtin_nontemporal_store(v, &out[idx]);   // streamed, never re-read
      }
    }
  }
}

extern "C" void kernel_launch(void* const* d_in, const int* in_sizes, int n_in,
                              void* d_out, int out_size, void* d_ws, size_t ws_size,
                              hipStream_t stream) {
  const float* left  = (const float*)d_in[0];
  const float* right = (const float*)d_in[1];
  // d_in[2] is max_disparity == 48, a structural constant baked into the kernel.
  float* out = (float*)d_out;

  dim3 grid(kW / 16 / 4, kH, 4);   // (5, 160, 4); 4 waves per block, 1 w-tile each
  dim3 block(128, 1, 1);
  hipLaunchKernelGGL(corr_volume_wmma, grid, block, 0, stream, left, right, out);
}